// CustomSwinEncoder_19344532701407
// MI455X (gfx1250) — compile-verified
//
#include <hip/hip_runtime.h>
#include <hip/hip_bf16.h>
#include <cstddef>

// ---------------------------------------------------------------------------
// CDNA5 (gfx1250) Swin encoder forward.
// All dense GEMMs run on V_WMMA_F32_16X16X4_F32 (fp32 matrix pipe, wave32).
// LDS tiles are laid out so every WMMA operand pair is a contiguous, 8B-aligned
// ds_load_b64 into an even VGPR pair (no v_mov repacking between WMMAs).
// ---------------------------------------------------------------------------

typedef __attribute__((ext_vector_type(2))) float v2f;
typedef __attribute__((ext_vector_type(8))) float v8f;

#define WMMA_F32(a, b, c) \
  __builtin_amdgcn_wmma_f32_16x16x4_f32(false, (a), false, (b), (short)0, (c), false, false)

static inline int ceil_div_i(long long a, long long b) { return (int)((a + b - 1) / b); }

__device__ __forceinline__ v2f ld2_lds(const float* p) { return *(const v2f*)p; }
__device__ __forceinline__ v2f ld2_g(const float* p)   { return *(const v2f*)p; }

// ======================= generic fp32 WMMA GEMM ============================
// C[M,N] = A[M,K] @ W[K,N] (+ bias[N]).  Macro tile 128x32, 8 waves/block,
// each wave owns a 16(M)x32(N) strip = two 16x16 WMMA accumulators.
// Requires: M % 128 == 0, N % 32 == 0, K % 16 == 0 (true for every GEMM here).
__global__ __launch_bounds__(256) void gemm_wmma_f32_kernel(
    const float* __restrict__ A, const float* __restrict__ W,
    const float* __restrict__ bias, float* __restrict__ C,
    int M, int N, int K) {
  __shared__ float As[128][18];   // even-pad: 8B-aligned rows, conflict-free
  __shared__ float Bt[32][18];    // B tile TRANSPOSED: Bt[n][k]
  const int tid  = threadIdx.x;
  const int wave = tid >> 5, lane = tid & 31;
  const int m0 = blockIdx.y * 128, n0 = blockIdx.x * 32;
  const int mrow = lane & 15;
  const int half = lane >> 4;
  const int kb   = half << 1;     // K sub-offset per WMMA fp32 operand layout
  v8f acc0 = {}; v8f acc1 = {};
  for (int k0 = 0; k0 < K; k0 += 16) {
    if (k0 + 16 < K) {            // prefetch next K-chunk (global_prefetch_b8)
      __builtin_prefetch(&A[(size_t)(m0 + (tid & 127)) * K + (k0 + 16) + ((tid >> 7) << 3)], 0, 1);
      if (tid < 16) __builtin_prefetch(&W[(size_t)(k0 + 16 + tid) * N + n0], 0, 1);
    }
    #pragma unroll
    for (int i = 0; i < 8; ++i) {          // 128x16 A tile, 2048 elems
      int idx = tid + (i << 8);
      int r = idx >> 4, c = idx & 15;
      As[r][c] = A[(size_t)(m0 + r) * K + (k0 + c)];
    }
    #pragma unroll
    for (int i = 0; i < 2; ++i) {          // 16x32 B tile, coalesced read,
      int idx = tid + (i << 8);            // transposed conflict-free LDS write
      int r = idx >> 5, c = idx & 31;
      Bt[c][r] = W[(size_t)(k0 + r) * N + (n0 + c)];
    }
    __syncthreads();
    const int am = (wave << 4) + mrow;
    #pragma unroll
    for (int kk = 0; kk < 16; kk += 4) {
      v2f a  = ld2_lds(&As[am][kk + kb]);          // A[m][k,k+1]
      v2f b0 = ld2_lds(&Bt[mrow][kk + kb]);        // B[k,k+1][n]
      v2f b1 = ld2_lds(&Bt[16 + mrow][kk + kb]);
      acc0 = WMMA_F32(a, b0, acc0);
      acc1 = WMMA_F32(a, b1, acc1);
    }
    __syncthreads();
  }
  const int col0 = n0 + mrow, col1 = col0 + 16;
  const float bv0 = bias ? bias[col0] : 0.f;
  const float bv1 = bias ? bias[col1] : 0.f;
  #pragma unroll
  for (int r = 0; r < 8; ++r) {   // C/D layout: lane<16 -> M=r, lane>=16 -> M=r+8
    int row = m0 + (wave << 4) + r + (half << 3);
    C[(size_t)row * N + col0] = acc0[r] + bv0;
    C[(size_t)row * N + col1] = acc1[r] + bv1;
  }
}

// ======================= window attention (WMMA) ===========================
__device__ __forceinline__ int swin_region(int x, int H) {
  return (x < H - 4) ? 0 : ((x < H - 2) ? 1 : 2);
}

// S = (Q K^T) * dh^-0.5 + rel_bias (+ shift mask).  One wave per (window,head).
__global__ __launch_bounds__(256) void attn_scores_kernel(
    const float* __restrict__ qkv, const float* __restrict__ bias_table,
    float* __restrict__ attn, int nwin_total, int nW, int nh, int C, int H, int shift) {
  const int wv = blockIdx.x * (blockDim.x >> 5) + (threadIdx.x >> 5);
  if (wv >= nwin_total * nh) return;            // wave-uniform guard (EXEC stays all-1)
  const int w = wv / nh, h = wv % nh;
  const int lane = threadIdx.x & 31;
  const int mrow = lane & 15, half = lane >> 4, kb = half << 1;
  const int C3 = 3 * C;
  const float* Q  = qkv + (size_t)w * 16 * C3 + h * 32;
  const float* Kp = Q + C;
  v8f acc = {};
  #pragma unroll
  for (int kk = 0; kk < 32; kk += 4) {          // dh = 32; operands are 8B pairs
    v2f a = ld2_g(Q  + (size_t)mrow * C3 + kk + kb);
    v2f b = ld2_g(Kp + (size_t)mrow * C3 + kk + kb);
    acc = WMMA_F32(a, b, acc);
  }
  const float scale = 0.17677669529663687f;     // 32^-0.5
  const int wi = w % nW, wpl = H / 4;
  const int wr = (wi / wpl) * 4, wc = (wi % wpl) * 4;
  const int nc = mrow;                          // key column
  int labn = 0;
  if (shift) {
    labn = swin_region(wr + (nc >> 2), H) * 3 + swin_region(wc + (nc & 3), H);
  }
  float* out = attn + (size_t)wv * 256;
  #pragma unroll
  for (int r = 0; r < 8; ++r) {
    const int m = r + (half << 3);              // query row
    const int dr = (m >> 2) - (nc >> 2) + 3, dc = (m & 3) - (nc & 3) + 3;
    float v = acc[r] * scale + bias_table[(dr * 7 + dc) * nh + h];
    if (shift) {
      int labm = swin_region(wr + (m >> 2), H) * 3 + swin_region(wc + (m & 3), H);
      if (labm != labn) v += -100.0f;
    }
    out[m * 16 + nc] = v;
  }
}

__global__ __launch_bounds__(256) void softmax16_kernel(float* __restrict__ a,
                                                        unsigned long long rows) {
  unsigned long long i = (unsigned long long)blockIdx.x * blockDim.x + threadIdx.x;
  if (i >= rows) return;
  float* p = a + i * 16;
  float mx = p[0];
  #pragma unroll
  for (int j = 1; j < 16; ++j) mx = fmaxf(mx, p[j]);
  float s = 0.f;
  #pragma unroll
  for (int j = 0; j < 16; ++j) { float e = expf(p[j] - mx); p[j] = e; s += e; }
  const float inv = 1.f / s;
  #pragma unroll
  for (int j = 0; j < 16; ++j) p[j] *= inv;
}

// out[w,t, h*32+d] = P[16x16] @ V[16x32].  One wave per (window,head).
__global__ __launch_bounds__(256) void attn_v_kernel(
    const float* __restrict__ attn, const float* __restrict__ qkv,
    float* __restrict__ out, int nwin_total, int nh, int C) {
  const int wv = blockIdx.x * (blockDim.x >> 5) + (threadIdx.x >> 5);
  if (wv >= nwin_total * nh) return;
  const int w = wv / nh, h = wv % nh;
  const int lane = threadIdx.x & 31;
  const int mrow = lane & 15, half = lane >> 4, kb = half << 1;
  const int C3 = 3 * C;
  const float* P = attn + (size_t)wv * 256;
  const float* V = qkv + (size_t)w * 16 * C3 + 2 * C + h * 32;
  v8f acc0 = {}; v8f acc1 = {};
  #pragma unroll
  for (int kk = 0; kk < 16; kk += 4) {
    v2f a = ld2_g(P + mrow * 16 + kk + kb);     // contiguous 8B pair
    v2f b0, b1;                                  // V rows are C3 apart: scalar
    b0.x = V[(size_t)(kk + kb) * C3 + mrow];      b0.y = V[(size_t)(kk + kb + 1) * C3 + mrow];
    b1.x = V[(size_t)(kk + kb) * C3 + 16 + mrow]; b1.y = V[(size_t)(kk + kb + 1) * C3 + 16 + mrow];
    acc0 = WMMA_F32(a, b0, acc0);
    acc1 = WMMA_F32(a, b1, acc1);
  }
  #pragma unroll
  for (int r = 0; r < 8; ++r) {
    const int row = r + (half << 3);
    float* o = out + ((size_t)w * 16 + row) * C + h * 32;
    o[mrow]      = acc0[r];
    o[mrow + 16] = acc1[r];
  }
}

// ======================= LN / elementwise / layout =========================
__global__ __launch_bounds__(256) void layernorm_kernel(
    const float* __restrict__ in, float* __restrict__ out,
    const float* __restrict__ g, const float* __restrict__ b, int rows, int C) {
  const int row = blockIdx.x * (blockDim.x >> 5) + (threadIdx.x >> 5);
  if (row >= rows) return;
  const int lane = threadIdx.x & 31;
  const float* p = in + (size_t)row * C;
  float s = 0.f, s2 = 0.f;
  for (int c = lane; c < C; c += 32) { float v = p[c]; s += v; s2 += v * v; }
  #pragma unroll
  for (int off = 16; off > 0; off >>= 1) {
    s  += __shfl_xor(s,  off, 32);
    s2 += __shfl_xor(s2, off, 32);
  }
  const float mean = s / C;
  const float inv  = rsqrtf(s2 / C - mean * mean + 1e-5f);
  float* q = out + (size_t)row * C;
  for (int c = lane; c < C; c += 32) {
    float v = (p[c] - mean) * inv;
    q[c] = g ? (v * g[c] + b[c]) : v;
  }
}

__global__ __launch_bounds__(256) void gelu_kernel(float* __restrict__ a,
                                                   unsigned long long n) {
  unsigned long long i = (unsigned long long)blockIdx.x * blockDim.x + threadIdx.x;
  if (i >= n) return;
  float v = a[i];
  a[i] = 0.5f * v * (1.0f + erff(v * 0.7071067811865476f));
}

__global__ __launch_bounds__(256) void relu_kernel(float* __restrict__ a,
                                                   unsigned long long n) {
  unsigned long long i = (unsigned long long)blockIdx.x * blockDim.x + threadIdx.x;
  if (i >= n) return;
  a[i] = fmaxf(a[i], 0.f);
}

__global__ __launch_bounds__(256) void add_kernel(float* __restrict__ x,
                                                  const float* __restrict__ y,
                                                  unsigned long long n) {
  unsigned long long i = (unsigned long long)blockIdx.x * blockDim.x + threadIdx.x;
  if (i >= n) return;
  x[i] += y[i];
}

// NCHW image -> patch matrix [B*32*32, 48], feature = (py*4+px)*3 + c
__global__ __launch_bounds__(256) void patchify_kernel(const float* __restrict__ x,
                                                       float* __restrict__ out,
                                                       unsigned long long n) {
  unsigned long long i = (unsigned long long)blockIdx.x * blockDim.x + threadIdx.x;
  if (i >= n) return;
  int f = (int)(i % 48); unsigned long long t = i / 48;
  int j = (int)(t % 32); t /= 32;
  int r = (int)(t % 32); int b = (int)(t / 32);
  int c = f % 3, pp = f / 3, py = pp >> 2, px = pp & 3;
  out[i] = x[(((size_t)b * 3 + c) * 128 + (r * 4 + py)) * 128 + (j * 4 + px)];
}

// [B,H,H,C] -> windows [B*nW, 16, C], with optional cyclic shift by -2
__global__ __launch_bounds__(256) void window_partition_kernel(
    const float* __restrict__ in, float* __restrict__ out,
    int H, int C, int shift, unsigned long long n) {
  unsigned long long i = (unsigned long long)blockIdx.x * blockDim.x + threadIdx.x;
  if (i >= n) return;
  int c = (int)(i % C); unsigned long long r = i / C;
  int t = (int)(r % 16); unsigned long long w = r / 16;
  const int wpl = H / 4, nW = wpl * wpl;
  int wi = (int)(w % nW); int b = (int)(w / nW);
  int hh = (wi / wpl) * 4 + (t >> 2);
  int ww = (wi % wpl) * 4 + (t & 3);
  if (shift) { hh = (hh + 2) % H; ww = (ww + 2) % H; }
  out[i] = in[(((size_t)b * H + hh) * H + ww) * C + c];
}

// x[B,H,H,C] += un-partition(win) with reverse shift (+2)
__global__ __launch_bounds__(256) void window_merge_add_kernel(
    float* __restrict__ x, const float* __restrict__ win,
    int H, int C, int shift, unsigned long long n) {
  unsigned long long i = (unsigned long long)blockIdx.x * blockDim.x + threadIdx.x;
  if (i >= n) return;
  int c = (int)(i % C); unsigned long long r = i / C;
  int ww = (int)(r % H); r /= H;
  int hh = (int)(r % H); int b = (int)(r / H);
  int hs = shift ? (hh + H - 2) % H : hh;
  int ws = shift ? (ww + H - 2) % H : ww;
  const int wpl = H / 4, nW = wpl * wpl;
  int wi = (hs >> 2) * wpl + (ws >> 2);
  int t  = (hs & 3) * 4 + (ws & 3);
  x[i] += win[(((size_t)b * nW + wi) * 16 + t) * C + c];
}

// 2x2 neighborhood concat: out[b,i,j,q*C+c], q order (0,0),(1,0),(0,1),(1,1)
__global__ __launch_bounds__(256) void patch_merge_kernel(
    const float* __restrict__ x, float* __restrict__ out,
    int H, int C, unsigned long long n) {
  unsigned long long i = (unsigned long long)blockIdx.x * blockDim.x + threadIdx.x;
  if (i >= n) return;
  const int C4 = 4 * C, H2 = H / 2;
  int f = (int)(i % C4); unsigned long long r = i / C4;
  int j = (int)(r % H2); r /= H2;
  int ii = (int)(r % H2); int b = (int)(r / H2);
  int q = f / C, c = f % C;
  int dh = q & 1, dw = q >> 1;
  out[i] = x[(((size_t)b * H + (2 * ii + dh)) * H + (2 * j + dw)) * C + c];
}

__global__ __launch_bounds__(256) void meanpool_kernel(const float* __restrict__ in,
                                                       float* __restrict__ out,
                                                       int tokens, int C,
                                                       unsigned long long n) {
  unsigned long long i = (unsigned long long)blockIdx.x * blockDim.x + threadIdx.x;
  if (i >= n) return;
  int c = (int)(i % C); int b = (int)(i / C);
  float s = 0.f;
  for (int t = 0; t < tokens; ++t) s += in[((size_t)b * tokens + t) * C + c];
  out[i] = s / tokens;
}

// ============================ host orchestration ===========================
extern "C" void kernel_launch(void* const* d_in, const int* in_sizes, int n_in,
                              void* d_out, int out_size, void* d_ws, size_t ws_size,
                              hipStream_t stream) {
  struct BlkP { const float *n1_g,*n1_b,*qkv_w,*qkv_b,*proj_w,*proj_b,*bias,
                            *n2_g,*n2_b,*w1,*b1,*w2,*b2; };
  struct MrgP { const float *g,*b,*w; };

  auto F = [&](int i) { return (const float*)d_in[i]; };
  const int X_ELEMS = 256 * 3 * 128 * 128;  // 12,582,912

  BlkP blk[8]; MrgP mrg[2];
  const float *x, *pe_w, *pe_b, *norm_g, *norm_b, *pj_w1, *pj_b1, *pj_w2, *pj_b2;

  const bool xFirst = (n_in > 0 && in_sizes[0] == X_ELEMS);
  const int  base   = xFirst ? 1 : 0;
  const bool sortedP = (in_sizes[base] == 147);   // alphabetical pytree: bias_table first

  int i = base;
  if (!sortedP) {                                  // dict insertion order
    pe_w = F(i++); pe_b = F(i++);
    for (int k = 0; k < 8; ++k) {
      blk[k].n1_g = F(i++); blk[k].n1_b = F(i++);
      blk[k].qkv_w = F(i++); blk[k].qkv_b = F(i++);
      blk[k].proj_w = F(i++); blk[k].proj_b = F(i++);
      blk[k].bias = F(i++);
      blk[k].n2_g = F(i++); blk[k].n2_b = F(i++);
      blk[k].w1 = F(i++); blk[k].b1 = F(i++);
      blk[k].w2 = F(i++); blk[k].b2 = F(i++);
    }
    for (int m = 0; m < 2; ++m) { mrg[m].g = F(i++); mrg[m].b = F(i++); mrg[m].w = F(i++); }
    norm_g = F(i++); norm_b = F(i++);
    pj_w1 = F(i++); pj_b1 = F(i++); pj_w2 = F(i++); pj_b2 = F(i++);
  } else {                                         // jax-sorted keys
    for (int k = 0; k < 8; ++k) {
      blk[k].bias = F(i++);
      blk[k].b1 = F(i++); blk[k].b2 = F(i++);
      blk[k].w1 = F(i++); blk[k].w2 = F(i++);
      blk[k].n1_b = F(i++); blk[k].n1_g = F(i++);
      blk[k].n2_b = F(i++); blk[k].n2_g = F(i++);
      blk[k].proj_b = F(i++); blk[k].proj_w = F(i++);
      blk[k].qkv_b = F(i++); blk[k].qkv_w = F(i++);
    }
    for (int m = 0; m < 2; ++m) { mrg[m].b = F(i++); mrg[m].g = F(i++); mrg[m].w = F(i++); }
    norm_b = F(i++); norm_g = F(i++);
    pe_b = F(i++); pe_w = F(i++);
    pj_b1 = F(i++); pj_b2 = F(i++); pj_w1 = F(i++); pj_w2 = F(i++);
  }
  x = xFirst ? F(0) : F(i);

  // --- workspace bump allocation (floats); S = max activation plane -------
  const size_t S = 25165824ULL;                 // 256*1024*96
  float* ws  = (float*)d_ws;
  float* X   = ws;                              // residual stream
  float* Y   = ws + S;                          // LN output
  float* Wb  = ws + 2 * S;                      // windows / scratch
  float* QKV = ws + 3 * S;                      // [rows, 3C]  (3*S)
  float* ATT = ws + 6 * S;                      // attention logits
  float* Hb  = ws + 7 * S;                      // attention output
  float* Y2  = ws + 8 * S;                      // proj / mlp output
  float* Mb  = ws + 3 * S;                      // MLP hidden (aliases dead QKV+ATT)
  (void)ws_size; (void)n_in; (void)out_size;

  auto gemm = [&](const float* A, const float* W, const float* bias, float* C,
                  int M, int N, int K) {
    dim3 grid(N / 32, M / 128);
    gemm_wmma_f32_kernel<<<grid, dim3(256), 0, stream>>>(A, W, bias, C, M, N, K);
  };
  auto ln = [&](const float* in, float* out, const float* g, const float* b,
                int rows, int C) {
    layernorm_kernel<<<ceil_div_i(rows, 8), 256, 0, stream>>>(in, out, g, b, rows, C);
  };
  auto ew = [&](unsigned long long n) { return ceil_div_i((long long)n, 256); };

  const int B = 256;
  // --- patch embed -------------------------------------------------------
  {
    unsigned long long n = (unsigned long long)B * 1024 * 48;
    patchify_kernel<<<ew(n), 256, 0, stream>>>(x, Wb, n);
    gemm(Wb, pe_w, pe_b, X, B * 1024, 96, 48);
    ln(X, X, nullptr, nullptr, B * 1024, 96);   // F.layer_norm without affine
  }

  const int Hs[3] = {32, 16, 8};
  const int Cs[3] = {96, 192, 384};
  const int NHs[3] = {3, 6, 12};
  const int depths[3] = {2, 2, 4};

  int bi = 0;
  for (int s = 0; s < 3; ++s) {
    const int H = Hs[s], C = Cs[s], nh = NHs[s];
    const int rows = B * H * H;
    const int nW = (H / 4) * (H / 4);
    const int nwin = B * nW;
    const unsigned long long nelem = (unsigned long long)rows * C;

    for (int j = 0; j < depths[s]; ++j) {
      const BlkP& p = blk[bi++];
      const int shift = j & 1;
      // --- window attention ---
      ln(X, Y, p.n1_g, p.n1_b, rows, C);
      window_partition_kernel<<<ew(nelem), 256, 0, stream>>>(Y, Wb, H, C, shift, nelem);
      gemm(Wb, p.qkv_w, p.qkv_b, QKV, rows, 3 * C, C);
      attn_scores_kernel<<<ceil_div_i((long long)nwin * nh, 8), 256, 0, stream>>>(
          QKV, p.bias, ATT, nwin, nW, nh, C, H, shift);
      softmax16_kernel<<<ew((unsigned long long)nwin * nh * 16), 256, 0, stream>>>(
          ATT, (unsigned long long)nwin * nh * 16);
      attn_v_kernel<<<ceil_div_i((long long)nwin * nh, 8), 256, 0, stream>>>(
          ATT, QKV, Hb, nwin, nh, C);
      gemm(Hb, p.proj_w, p.proj_b, Y2, rows, C, C);
      window_merge_add_kernel<<<ew(nelem), 256, 0, stream>>>(X, Y2, H, C, shift, nelem);
      // --- MLP ---
      ln(X, Y, p.n2_g, p.n2_b, rows, C);
      gemm(Y, p.w1, p.b1, Mb, rows, 4 * C, C);
      gelu_kernel<<<ew(nelem * 4), 256, 0, stream>>>(Mb, nelem * 4);
      gemm(Mb, p.w2, p.b2, Y2, rows, C, 4 * C);
      add_kernel<<<ew(nelem), 256, 0, stream>>>(X, Y2, nelem);
    }
    if (s < 2) {  // patch merging: [B,H,H,C] -> [B,H/2,H/2,4C] -> LN -> @W -> 2C
      patch_merge_kernel<<<ew(nelem), 256, 0, stream>>>(X, Y, H, C, nelem);
      ln(Y, Wb, mrg[s].g, mrg[s].b, rows / 4, 4 * C);
      gemm(Wb, mrg[s].w, nullptr, X, rows / 4, 2 * C, 4 * C);
    }
  }

  // --- final LN, mean pool -> h, projection head -> z --------------------
  float* out = (float*)d_out;           // [h: 256*384][z: 256*256]
  ln(X, Y, norm_g, norm_b, B * 64, 384);
  {
    unsigned long long n = (unsigned long long)B * 384;
    meanpool_kernel<<<ew(n), 256, 0, stream>>>(Y, out, 64, 384, n);
  }
  gemm(out, pj_w1, pj_b1, Y, B, 384, 384);
  relu_kernel<<<ew((unsigned long long)B * 384), 256, 0, stream>>>(
      Y, (unsigned long long)B * 384);
  gemm(Y, pj_w2, pj_b2, out + (size_t)B * 384, B, 256, 384);
}